// ANEGemmaAttention_66769561584221
// MI455X (gfx1250) — compile-verified
//
#include <hip/hip_runtime.h>
#include <hip/hip_bf16.h>
#include <math.h>

// ---------------- constants ----------------
#define HEAD_DIM   256
#define NUM_HEADS  8
#define HIDDEN     2048
#define MAX_SEQ    8192
#define L5_BASE    ((size_t)5 * MAX_SEQ * HEAD_DIM)   // layer 5, kv-head 0
#define QKV_ROWS   2560
#define SOFTCAP    50.0f
#define SCALING    0.0625f          // 256^-0.5

// workspace layout (float offsets)
#define QKV_OFF    0                       // 2560
#define QPAD_OFF   2560                    // 16*256 = 4096
#define KNEW_OFF   (QPAD_OFF + 4096)       // 256
#define VNEW_OFF   (KNEW_OFF + 256)        // 256
#define SCORES_OFF (VNEW_OFF + 256)        // 8*8192 = 65536
#define NPVCHUNK   64                      // 128 positions each
#define PART_OFF   (SCORES_OFF + 65536)    // 64*2048 = 131072
#define OATTN_OFF  (PART_OFF + 131072)     // 2048

typedef __attribute__((ext_vector_type(2))) float v2f;
typedef __attribute__((ext_vector_type(8))) float v8f;
typedef __attribute__((ext_vector_type(4))) int   v4i;

// --- async global->LDS copy (gfx1250) with guarded fallback ---
#ifndef HAVE_ASYNC_LDS
#if defined(__has_builtin)
#if __has_builtin(__builtin_amdgcn_global_load_async_to_lds_b128)
#define HAVE_ASYNC_LDS 1
#endif
#endif
#endif
#ifndef HAVE_ASYNC_LDS
#define HAVE_ASYNC_LDS 0
#endif

// ---------------- GEMV: y[row] = W[row,:] . x ----------------
// block = 256 threads (8 waves), one row per wave, float4 streams
__global__ void gemv_kernel(const float* __restrict__ W, const float* __restrict__ x,
                            float* __restrict__ y, int ncols) {
    int wave = threadIdx.x >> 5, lane = threadIdx.x & 31;
    int row  = blockIdx.x * 8 + wave;
    const float4* w4 = (const float4*)(W + (size_t)row * ncols);
    const float4* x4 = (const float4*)x;
    int n4 = ncols >> 2;
    float acc = 0.0f;
    for (int i = lane; i < n4; i += 32) {
        float4 a = w4[i], b = x4[i];
        acc += a.x * b.x + a.y * b.y + a.z * b.z + a.w * b.w;
    }
    for (int off = 16; off; off >>= 1) acc += __shfl_xor(acc, off, 32);
    if (lane == 0) y[row] = acc;
}

// ---------------- Q/K rmsnorm + RoPE ----------------
// 1 block x 288 threads (9 waves): waves 0-7 -> q heads, wave 8 -> k
__global__ void qk_prep_kernel(const float* __restrict__ cosv, const float* __restrict__ sinv,
                               const float* __restrict__ qnw, const float* __restrict__ knw,
                               float* __restrict__ ws) {
    __shared__ float tmp[9][HEAD_DIM];
    const float* qkv = ws + QKV_OFF;
    float* qpad = ws + QPAD_OFF;
    float* knew = ws + KNEW_OFF;
    float* vnew = ws + VNEW_OFF;
    int tid = threadIdx.x, wave = tid >> 5, lane = tid & 31;

    if (tid < HEAD_DIM) {
        vnew[tid] = qkv[(NUM_HEADS + 1) * HEAD_DIM + tid];   // v slice
        #pragma unroll
        for (int r = 8; r < 16; ++r) qpad[r * HEAD_DIM + tid] = 0.0f;  // pad heads
    }

    const float* x = qkv + wave * HEAD_DIM;       // wave 8 -> k slice (offset 2048)
    const float* w = (wave == 8) ? knw : qnw;

    float xs[8];
    float mx = 0.0f;
    #pragma unroll
    for (int i = 0; i < 8; ++i) {
        float v = x[lane + i * 32];
        xs[i] = v;
        mx = fmaxf(mx, fabsf(v));
    }
    for (int off = 16; off; off >>= 1) mx = fmaxf(mx, __shfl_xor(mx, off, 32));
    mx = fmaxf(mx, 5.9604644775390625e-8f);       // 2^-24
    float inv = 1.0f / mx;
    float ss = 0.0f;
    #pragma unroll
    for (int i = 0; i < 8; ++i) { float v = xs[i] * inv; xs[i] = v; ss += v * v; }
    for (int off = 16; off; off >>= 1) ss += __shfl_xor(ss, off, 32);
    float rs = rsqrtf(ss) * 16.0f;                // sqrt(HEAD_DIM) = 16
    #pragma unroll
    for (int i = 0; i < 8; ++i) {
        int d = lane + i * 32;
        tmp[wave][d] = rs * xs[i] * (1.0f + w[d]);
    }
    __syncthreads();
    #pragma unroll
    for (int i = 0; i < 8; ++i) {
        int d = lane + i * 32;
        float xv = tmp[wave][d];
        float xr = (d < 128) ? -tmp[wave][d + 128] : tmp[wave][d - 128];
        float o = xv * cosv[d] + xr * sinv[d];
        if (wave == 8) knew[d] = o;
        else           qpad[wave * HEAD_DIM + d] = o * SCALING;
    }
}

// ---------------- scores via V_WMMA_F32_16X16X4_F32 ----------------
// block = 128 threads (4 waves); each wave: one 16(head)x16(pos) tile.
// grid 128 blocks * 64 pos/block = 8192 positions.
// K tiles are staged into LDS with GLOBAL_LOAD_ASYNC_TO_LDS_B128 (coalesced,
// ASYNCcnt-tracked) when the toolchain exposes the builtin.
__global__ void scores_kernel(const float* __restrict__ kc, const float* __restrict__ mask,
                              const int* __restrict__ posp, float* __restrict__ ws) {
    const float* qpad  = ws + QPAD_OFF;
    const float* knew  = ws + KNEW_OFF;
    float* scores      = ws + SCORES_OFF;
    __shared__ float qs[16 * HEAD_DIM];           // 16 KB
#if HAVE_ASYNC_LDS
    __shared__ float ks[4][16 * HEAD_DIM];        // 64 KB, one tile per wave
#endif
    int tid = threadIdx.x;
    for (int i = tid; i < 16 * HEAD_DIM; i += 128) qs[i] = qpad[i];
    __syncthreads();

    int wave = tid >> 5, lane = tid & 31;
    int tile_base = blockIdx.x * 64 + wave * 16;
    int n   = lane & 15;                          // position / A-row index within tile
    int hi  = lane >> 4;                          // half select (K pair)
    int pos = tile_base + n;
    float mval = mask[pos];
    bool allmasked = __all(mval < -1.0e8f);

    v8f c = {};
    if (!allmasked) {
        int POS = posp[0];
        const float* arow = &qs[n * HEAD_DIM];    // A: M = lane&15
        int cb = 2 * hi;                          // lanes 16-31 carry K=+2,+3

#if HAVE_ASYNC_LDS
        // ---- stage 16x256 fp32 K tile into LDS (16 KB) with async b128 copies
        // 1024 16-byte chunks; 32 lanes x 32 issues; each issue = 512 B
        // contiguous within a K row -> fully coalesced.
        const float* ksrc = kc + L5_BASE + (size_t)tile_base * HEAD_DIM;
        #pragma unroll 4
        for (int j = 0; j < 32; ++j) {
            int e    = j * 32 + lane;             // 16B-chunk index in tile
            int row  = e >> 6;                    // 64 chunks per 256-float row
            int col4 = e & 63;
            const float* src = ksrc + (size_t)row * HEAD_DIM + col4 * 4;
            float* dst = &ks[wave][row * HEAD_DIM + col4 * 4];
            __builtin_amdgcn_global_load_async_to_lds_b128(
                (v4i*)(void*)src, (v4i*)(void*)dst, 0, 0);
        }
        asm volatile("s_wait_asynccnt 0x0" ::: "memory");
        // patch in the freshly-computed key row (wave-uniform branch)
        if (POS >= tile_base && POS < tile_base + 16) {
            int r = POS - tile_base;
            for (int i = lane; i < HEAD_DIM; i += 32) ks[wave][r * HEAD_DIM + i] = knew[i];
        }
        const float* krow = &ks[wave][n * HEAD_DIM];   // B: N = lane&15, from LDS
#else
        const float* krow = (pos == POS) ? knew
                            : (kc + L5_BASE + (size_t)pos * HEAD_DIM);
#endif
        #pragma unroll 8
        for (int k0 = 0; k0 < HEAD_DIM; k0 += 4) {
            v2f a, b;
            a.x = arow[k0 + cb];  a.y = arow[k0 + cb + 1];
            b.x = krow[k0 + cb];  b.y = krow[k0 + cb + 1];
            c = __builtin_amdgcn_wmma_f32_16x16x4_f32(
                    false, a, false, b, (short)0, c, false, false);
        }
    }
    // C layout: VGPR r -> head r (lanes 0-15) / head r+8 (lanes 16-31), N = lane&15
    #pragma unroll
    for (int r = 0; r < 8; ++r) {
        float s = SOFTCAP * tanhf(c[r] * (1.0f / SOFTCAP)) + mval;
        if (hi == 0) scores[r * MAX_SEQ + pos] = s;   // heads 8-15 are padding
    }
}

// ---------------- per-head softmax over 8192 ----------------
__global__ void softmax_kernel(float* __restrict__ ws) {
    float* row = ws + SCORES_OFF + (size_t)blockIdx.x * MAX_SEQ;
    __shared__ float red[8];
    int tid = threadIdx.x, wave = tid >> 5, lane = tid & 31;

    float m = -3.0e38f;
    for (int i = tid; i < MAX_SEQ; i += 256) m = fmaxf(m, row[i]);
    for (int off = 16; off; off >>= 1) m = fmaxf(m, __shfl_xor(m, off, 32));
    if (lane == 0) red[wave] = m;
    __syncthreads();
    m = red[0];
    #pragma unroll
    for (int w = 1; w < 8; ++w) m = fmaxf(m, red[w]);
    __syncthreads();

    float s = 0.0f;
    for (int i = tid; i < MAX_SEQ; i += 256) {
        float e = __expf(row[i] - m);
        row[i] = e;
        s += e;
    }
    for (int off = 16; off; off >>= 1) s += __shfl_xor(s, off, 32);
    if (lane == 0) red[wave] = s;
    __syncthreads();
    s = 0.0f;
    #pragma unroll
    for (int w = 0; w < 8; ++w) s += red[w];
    float invs = 1.0f / s;
    for (int i = tid; i < MAX_SEQ; i += 256) row[i] *= invs;
}

// ---------------- probs @ V partials, 128 positions per block ----------------
__global__ void pv_partial_kernel(const float* __restrict__ vc, const int* __restrict__ posp,
                                  float* __restrict__ ws) {
    const float* scores = ws + SCORES_OFF;        // now probs
    const float* vnew   = ws + VNEW_OFF;
    float* part         = ws + PART_OFF;
    __shared__ float lp[8][128];
    __shared__ int anyflag;
    int tid = threadIdx.x;                        // 256 threads; tid == d index
    int base = blockIdx.x * 128;
    if (tid == 0) anyflag = 0;
    __syncthreads();

    int nz = 0;
    for (int i = tid; i < 8 * 128; i += 256) {
        int h = i >> 7, s = i & 127;
        float p = scores[(size_t)h * MAX_SEQ + base + s];
        lp[h][s] = p;
        nz |= (p != 0.0f);
    }
    if (nz) atomicOr(&anyflag, 1);
    __syncthreads();

    float acc[8] = {0, 0, 0, 0, 0, 0, 0, 0};
    if (anyflag) {
        int POS = posp[0];
        for (int s = 0; s < 128; ++s) {
            int pos = base + s;
            const float* vrow = (pos == POS) ? vnew
                                : (vc + L5_BASE + (size_t)pos * HEAD_DIM);
            float v = vrow[tid];
            #pragma unroll
            for (int h = 0; h < 8; ++h) acc[h] += lp[h][s] * v;
        }
    }
    #pragma unroll
    for (int h = 0; h < 8; ++h)
        part[(size_t)blockIdx.x * 2048 + h * HEAD_DIM + tid] = acc[h];
}

// ---------------- reduce partials -> out_attn[2048] ----------------
__global__ void reduce_kernel(float* __restrict__ ws) {
    int c = blockIdx.x * 256 + threadIdx.x;       // 8 blocks -> 2048
    float a = 0.0f;
    for (int p = 0; p < NPVCHUNK; ++p) a += ws[PART_OFF + (size_t)p * 2048 + c];
    ws[OATTN_OFF + c] = a;
}

// ---------------- launch ----------------
extern "C" void kernel_launch(void* const* d_in, const int* in_sizes, int n_in,
                              void* d_out, int out_size, void* d_ws, size_t ws_size,
                              hipStream_t stream) {
    const float* hs   = (const float*)d_in[0];
    const float* cosv = (const float*)d_in[1];
    const float* sinv = (const float*)d_in[2];
    const int*   posp = (const int*)  d_in[3];
    const float* kc   = (const float*)d_in[4];
    const float* vc   = (const float*)d_in[5];
    const float* mask = (const float*)d_in[6];
    const float* qkvw = (const float*)d_in[7];
    const float* ow   = (const float*)d_in[8];
    const float* qnw  = (const float*)d_in[9];
    const float* knw  = (const float*)d_in[10];
    float* out = (float*)d_out;
    float* ws  = (float*)d_ws;

    (void)in_sizes; (void)n_in; (void)out_size; (void)ws_size;

    // 1) qkv = qkv_w @ hidden  (2560 rows)
    gemv_kernel<<<QKV_ROWS / 8, 256, 0, stream>>>(qkvw, hs, ws + QKV_OFF, HIDDEN);
    // 2) rmsnorm + rope + scaling; v copy; q pad
    qk_prep_kernel<<<1, 288, 0, stream>>>(cosv, sinv, qnw, knw, ws);
    // 3) scores via fp32 WMMA + softcap + mask (K tiles async-staged to LDS)
    scores_kernel<<<MAX_SEQ / 64, 128, 0, stream>>>(kc, mask, posp, ws);
    // 4) softmax per head
    softmax_kernel<<<NUM_HEADS, 256, 0, stream>>>(ws);
    // 5) probs @ V partials
    pv_partial_kernel<<<NPVCHUNK, 256, 0, stream>>>(vc, posp, ws);
    // 6) reduce partials
    reduce_kernel<<<8, 256, 0, stream>>>(ws);
    // 7) final projection
    gemv_kernel<<<HIDDEN / 8, 256, 0, stream>>>(ow, ws + OATTN_OFF, out, HIDDEN);
}